// GCN_42623255445707
// MI455X (gfx1250) — compile-verified
//
#include <hip/hip_runtime.h>
#include <hip/hip_bf16.h>
#include <math.h>

// ---------------------------------------------------------------------------
// GCN forward for MI455X (gfx1250, wave32).
//  - GEMMs: V_WMMA_F32_16X16X4_F32, weights pre-transposed/padded so both A and
//    B lane fragments are contiguous float2 (global_load_b64), no guards in loop.
//  - Aggregation: one-time CSR build (by dst), then per-layer float4 gather-
//    accumulate fused with self-loop term + bias + ReLU. No per-layer atomics.
// ---------------------------------------------------------------------------

typedef float v2f_t __attribute__((ext_vector_type(2)));
typedef float v8f_t __attribute__((ext_vector_type(8)));

// ---------------- small utility kernels ----------------

__global__ void fill_int_kernel(int* __restrict__ p, int v, int n) {
    int i = blockIdx.x * blockDim.x + threadIdx.x;
    if (i < n) p[i] = v;
}

__global__ void copy_int_kernel(const int* __restrict__ a, int* __restrict__ b, int n) {
    int i = blockIdx.x * blockDim.x + threadIdx.x;
    if (i < n) b[i] = a[i];
}

__global__ void deg_count_kernel(const int* __restrict__ dst, int* __restrict__ cnt, int E) {
    int e = blockIdx.x * blockDim.x + threadIdx.x;
    if (e < E) atomicAdd(&cnt[dst[e]], 1);
}

// dinv[i] = rsqrt(1 + in_degree(i))   (self-loop included)
__global__ void dinv_kernel(const int* __restrict__ cnt, float* __restrict__ dinv, int n) {
    int i = blockIdx.x * blockDim.x + threadIdx.x;
    if (i < n) dinv[i] = rsqrtf((float)cnt[i] + 1.0f);
}

// single-block Hillis-Steele scan with carry: rowstart = exclusive_scan(cnt)
__global__ __launch_bounds__(1024) void scan_kernel(const int* __restrict__ cnt,
                                                    int* __restrict__ rowstart, int n) {
    __shared__ int buf[1024];
    __shared__ int carry;
    int t = (int)threadIdx.x;
    if (t == 0) carry = 0;
    __syncthreads();
    for (int base = 0; base < n; base += 1024) {
        int i = base + t;
        int v = (i < n) ? cnt[i] : 0;
        buf[t] = v;
        __syncthreads();
        for (int off = 1; off < 1024; off <<= 1) {
            int add = (t >= off) ? buf[t - off] : 0;
            __syncthreads();
            buf[t] += add;
            __syncthreads();
        }
        int excl = carry + buf[t] - v;
        if (i < n) rowstart[i] = excl;
        __syncthreads();
        if (t == 0) carry += buf[1023];
        __syncthreads();
    }
    if (t == 0) rowstart[n] = carry;
}

// place each edge into CSR bucket of its dst; pack (src, norm) as int2
__global__ void place_kernel(const int* __restrict__ src, const int* __restrict__ dst,
                             const float* __restrict__ dinv, int* __restrict__ cursor,
                             int2* __restrict__ cpack, int E) {
    int e = blockIdx.x * blockDim.x + threadIdx.x;
    if (e >= E) return;
    int s = src[e];
    int d = dst[e];
    float nrm = dinv[s] * dinv[d];
    int pos = atomicAdd(&cursor[d], 1);
    int2 pk;
    pk.x = s;
    pk.y = __float_as_int(nrm);
    cpack[pos] = pk;
}

// Wt[col][k] = (col < Fout) ? W[k][col] : 0   ; Wt dims [FoutPad][K]
__global__ void transpose_pad_kernel(const float* __restrict__ W, float* __restrict__ Wt,
                                     int K, int Fout, int FoutPad) {
    int i = blockIdx.x * blockDim.x + threadIdx.x;
    int total = FoutPad * K;
    if (i >= total) return;
    int col = i / K;
    int k   = i - col * K;
    Wt[i] = (col < Fout) ? W[(size_t)k * Fout + col] : 0.0f;
}

// ---------------- WMMA fp32 GEMM: h = x @ W ----------------
// x: [M,K] row-major; Wt: [NT*16][K] (transposed, zero-padded weights).
// One wave -> 16-row stripe x NT 16-col tiles. All fragment loads are b64.
// A 16x4 f32: lane(0..15)=M, half hf -> K pair {2hf, 2hf+1}
// B 4x16 f32: lane(0..15)=N, half hf -> K pair {2hf, 2hf+1}
// D 16x16:    lane(0..15)=N, VGPR v at half hf -> M = v + 8*hf

template <int NT>
__global__ __launch_bounds__(256) void gemm_wmma_kernel(
    const float* __restrict__ x, const float* __restrict__ Wt,
    float* __restrict__ h, int M, int K, int Fout) {
    int wave   = (int)((blockIdx.x * blockDim.x + threadIdx.x) >> 5);
    int Mtiles = (M + 15) >> 4;
    if (wave >= Mtiles) return;  // wave-uniform: EXEC all-1s at every WMMA

    int lane = (int)(threadIdx.x & 31u);
    int hf   = lane >> 4;
    int lm   = lane & 15;

    int row = wave * 16 + lm;
    if (row >= M) row = M - 1;  // clamp (M % 16 == 0 here)
    const float* xrow = x + (size_t)row * K + 2 * hf;
    const float* wcol[NT];
#pragma unroll
    for (int t = 0; t < NT; ++t)
        wcol[t] = Wt + (size_t)(t * 16 + lm) * K + 2 * hf;

    v8f_t acc[NT];
#pragma unroll
    for (int t = 0; t < NT; ++t) acc[t] = {};

#pragma unroll 4
    for (int k0 = 0; k0 < K; k0 += 4) {
        v2f_t a = *(const v2f_t*)(xrow + k0);
#pragma unroll
        for (int t = 0; t < NT; ++t) {
            v2f_t b = *(const v2f_t*)(wcol[t] + k0);
            acc[t] = __builtin_amdgcn_wmma_f32_16x16x4_f32(
                false, a, false, b, (short)0, acc[t], false, false);
        }
    }

#pragma unroll
    for (int t = 0; t < NT; ++t) {
        int col = t * 16 + lm;
        if (col < Fout) {
#pragma unroll
            for (int v = 0; v < 8; ++v) {
                int r = wave * 16 + v + 8 * hf;
                if (r < M) h[(size_t)r * Fout + col] = acc[t][v];
            }
        }
    }
}

// ---------------- fused CSR aggregation (float4 vectorized) ----------------
// out[d][f] = [relu]( sum_j h[src_j][f]*norm_j + h[d][f]*dinv[d]^2 + bias[f] )
// F/4 threads per node, each owning a float4 slice (global_load_b128 gathers).

template <int F, bool RELU>
__global__ __launch_bounds__(256) void aggregate_kernel(
    const float* __restrict__ h, const int* __restrict__ rowstart,
    const int2* __restrict__ cpack, const float* __restrict__ dinv,
    const float* __restrict__ bias, float* __restrict__ out, int N) {
    constexpr int TPN = F / 4;                       // threads per node
    constexpr int SH  = (TPN == 16) ? 4 : 0;         // log2(TPN) for F=64 / F=4
    static_assert(TPN == 16 || TPN == 1, "F must be 64 or 4");

    int tid  = blockIdx.x * blockDim.x + threadIdx.x;
    int node = tid >> SH;
    int q    = tid & (TPN - 1);
    if (node >= N) return;
    int f0 = q * 4;

    int beg = rowstart[node];
    int end = rowstart[node + 1];
    float s = dinv[node];

    float4 a = *(const float4*)(h + (size_t)node * F + f0);
    float ss = s * s;
    float ax = a.x * ss, ay = a.y * ss, az = a.z * ss, aw = a.w * ss;

    for (int j = beg; j < end; ++j) {
        int2 pk = cpack[j];
        float w = __int_as_float(pk.y);
        float4 v = *(const float4*)(h + (size_t)pk.x * F + f0);
        ax += v.x * w;
        ay += v.y * w;
        az += v.z * w;
        aw += v.w * w;
    }

    float4 b4 = *(const float4*)(bias + f0);
    ax += b4.x; ay += b4.y; az += b4.z; aw += b4.w;
    if (RELU) {
        ax = fmaxf(ax, 0.0f); ay = fmaxf(ay, 0.0f);
        az = fmaxf(az, 0.0f); aw = fmaxf(aw, 0.0f);
    }
    float4 r;
    r.x = ax; r.y = ay; r.z = az; r.w = aw;
    *(float4*)(out + (size_t)node * F + f0) = r;
}

// ---------------- column softmax over node dim (axis=0), C=4 ----------------

__global__ __launch_bounds__(256) void softmax_col_kernel(
    const float* __restrict__ in, float* __restrict__ out, int N) {
    const int C = 4;
    int c = blockIdx.x;
    __shared__ float sred[256];
    int t = (int)threadIdx.x;

    float m = -3.402823466e38f;
    for (int i = t; i < N; i += 256) m = fmaxf(m, in[(size_t)i * C + c]);
    sred[t] = m;
    __syncthreads();
    for (int s = 128; s > 0; s >>= 1) {
        if (t < s) sred[t] = fmaxf(sred[t], sred[t + s]);
        __syncthreads();
    }
    m = sred[0];
    __syncthreads();

    float sum = 0.0f;
    for (int i = t; i < N; i += 256) sum += expf(in[(size_t)i * C + c] - m);
    sred[t] = sum;
    __syncthreads();
    for (int s = 128; s > 0; s >>= 1) {
        if (t < s) sred[t] += sred[t + s];
        __syncthreads();
    }
    float inv = 1.0f / sred[0];

    for (int i = t; i < N; i += 256)
        out[(size_t)i * C + c] = expf(in[(size_t)i * C + c] - m) * inv;
}

// ---------------------------------------------------------------------------

static inline int ceil_div_i(long long a, long long b) { return (int)((a + b - 1) / b); }

struct GcnCtx {
    const int* rowstart;
    const int2* cpack;
    const float* dinv;
    int N;
    hipStream_t stream;
};

template <int NT, int F, bool RELU>
static void run_layer_t(const GcnCtx& ctx, const float* x, int K, const float* Wt,
                        const float* bias, int Fout, float* hbuf, float* out) {
    int Mtiles = (ctx.N + 15) >> 4;
    gemm_wmma_kernel<NT><<<ceil_div_i(Mtiles, 8), 256, 0, ctx.stream>>>(
        x, Wt, hbuf, ctx.N, K, Fout);
    long long nthr = (long long)ctx.N * (F / 4);
    aggregate_kernel<F, RELU><<<ceil_div_i(nthr, 256), 256, 0, ctx.stream>>>(
        hbuf, ctx.rowstart, ctx.cpack, ctx.dinv, bias, out, ctx.N);
}

extern "C" void kernel_launch(void* const* d_in, const int* in_sizes, int n_in,
                              void* d_out, int out_size, void* d_ws,
                              size_t ws_size, hipStream_t stream) {
    const float* x0   = (const float*)d_in[0];  // [N,128]
    const int*   eidx = (const int*)d_in[1];    // [2,E]
    const float* W1   = (const float*)d_in[2];  // [128,64]
    const float* b1   = (const float*)d_in[3];
    const float* W2   = (const float*)d_in[4];  // [64,64]
    const float* b2   = (const float*)d_in[5];
    const float* W3   = (const float*)d_in[6];  // [64,4]
    const float* b3   = (const float*)d_in[7];

    const int N = in_sizes[0] / 128;
    const int E = in_sizes[1] / 2;
    const int* src = eidx;
    const int* dst = eidx + E;

    // workspace carve-up (4-byte units; every chunk offset stays 16B-aligned:
    // N and E are multiples of 4, rowstart padded to N+4)
    float* ws = (float*)d_ws;
    float* dinv     = ws;        ws += N;                 // [N]
    int*   rowstart = (int*)ws;  ws += N + 4;             // [N+1] (+3 pad)
    int*   cursor   = (int*)ws;  ws += N;                 // counts, then cursors
    int2*  cpack    = (int2*)ws; ws += (size_t)2 * E;     // (src, norm) per edge
    float* hbuf     = ws;        ws += (size_t)N * 64;    // h = x @ W
    float* xbuf     = ws;        ws += (size_t)N * 64;    // activations
    float* logits   = ws;        ws += (size_t)N * 4;     // pre-softmax
    float* Wt1      = ws;        ws += 64 * 128;          // [64][128]
    float* Wt2      = ws;        ws += 64 * 64;           // [64][64]
    float* Wt3p     = ws;        ws += 16 * 64;           // [16][64] zero-padded

    // ---- one-time graph preprocessing: degrees, dinv, CSR by dst -----------
    fill_int_kernel<<<ceil_div_i(N, 256), 256, 0, stream>>>(cursor, 0, N);
    deg_count_kernel<<<ceil_div_i(E, 256), 256, 0, stream>>>(dst, cursor, E);
    dinv_kernel<<<ceil_div_i(N, 256), 256, 0, stream>>>(cursor, dinv, N);
    scan_kernel<<<1, 1024, 0, stream>>>(cursor, rowstart, N);
    copy_int_kernel<<<ceil_div_i(N, 256), 256, 0, stream>>>(rowstart, cursor, N);
    place_kernel<<<ceil_div_i(E, 256), 256, 0, stream>>>(src, dst, dinv, cursor,
                                                         cpack, E);

    // ---- one-time weight transpose / pad -----------------------------------
    transpose_pad_kernel<<<ceil_div_i(64 * 128, 256), 256, 0, stream>>>(W1, Wt1, 128, 64, 64);
    transpose_pad_kernel<<<ceil_div_i(64 * 64, 256), 256, 0, stream>>>(W2, Wt2, 64, 64, 64);
    transpose_pad_kernel<<<ceil_div_i(16 * 64, 256), 256, 0, stream>>>(W3, Wt3p, 64, 4, 16);

    GcnCtx ctx{rowstart, cpack, dinv, N, stream};

    // ---- layer 1: 128 -> 64, ReLU ------------------------------------------
    run_layer_t<4, 64, true>(ctx, x0, 128, Wt1, b1, 64, hbuf, xbuf);

    // ---- layers 2..8: 64 -> 64 shared weights, ReLU ------------------------
    for (int l = 0; l < 7; ++l)
        run_layer_t<4, 64, true>(ctx, xbuf, 64, Wt2, b2, 64, hbuf, xbuf);

    // ---- layer 9: 64 -> 4, no activation -----------------------------------
    run_layer_t<1, 4, false>(ctx, xbuf, 64, Wt3p, b3, 4, hbuf, logits);

    // ---- softmax over node dimension (axis 0) ------------------------------
    softmax_col_kernel<<<4, 256, 0, stream>>>(logits, (float*)d_out, N);
}